// GAT_15994458210591
// MI455X (gfx1250) — compile-verified
//
#include <hip/hip_runtime.h>
#include <hip/hip_bf16.h>
#include <math.h>

typedef float v2f __attribute__((ext_vector_type(2)));
typedef float v8f __attribute__((ext_vector_type(8)));

#define N_NODES   65536
#define F_IN      128
#define N_EDGES   1048576
#define E_TOT     (N_EDGES + N_NODES)   /* edges + self loops = 1114112 */
#define HEADS     4
#define HID       64
#define HC        256                   /* HEADS*HID */
#define OUTC      512                   /* xs(256) ++ xd(256) */
#define NEG_SLOPE 0.2f
#define BN_EPS    1e-5f

/* ---- workspace layout (float offsets into d_ws) ---- */
#define SCALE_OFF  ((size_t)0)                               /* 128 f32: gamma*invstd       */
#define SHIFT_OFF  ((size_t)128)                             /* 128 f32: beta-mean*scale    */
#define BNSUM_OFF  ((size_t)256)                             /* 256 doubles (512 f32 slots) */
#define WT_OFF     ((size_t)768)                             /* wT2[(K/2)=64][N=512][2]     */
#define BIAS2_OFF  (WT_OFF + (size_t)F_IN * OUTC)            /* 512 f32                     */
#define XSD_OFF    ((size_t)131072)                          /* [N][512] xs|xd  (128 MB)    */
#define ACC_OFF    (XSD_OFF + (size_t)N_NODES * OUTC)        /* [N][256] head acc (64 MB)   */
#define SEGMAX_OFF (ACC_OFF + (size_t)N_NODES * HC)          /* [N][4] encoded-max (u32)    */
#define SEGSUM_OFF (SEGMAX_OFF + (size_t)N_NODES * HEADS)    /* [N][4] exp-sum              */
#define P_OFF      (SEGSUM_OFF + (size_t)N_NODES * HEADS)    /* [E_TOT][4] logits -> p      */

/* monotonic order-preserving f32 <-> u32 encoding for atomicMax */
__device__ __forceinline__ unsigned enc_f32(float f) {
    unsigned u = __float_as_uint(f);
    return (u & 0x80000000u) ? ~u : (u | 0x80000000u);
}
__device__ __forceinline__ float dec_f32(unsigned k) {
    return __uint_as_float((k & 0x80000000u) ? (k & 0x7FFFFFFFu) : ~k);
}

/* ---------------- init: zero all accumulated scratch (every call) -------- */
__global__ __launch_bounds__(256) void k_init(float* __restrict__ ws) {
    size_t t = (size_t)blockIdx.x * 256 + threadIdx.x;   /* grid covers N*HC */
    if (t < (size_t)N_NODES * HC) ws[ACC_OFF + t] = 0.0f;
    if (t < (size_t)N_NODES * HEADS) {
        ((unsigned*)(ws + SEGMAX_OFF))[t] = 0u;          /* < enc of any finite */
        ws[SEGSUM_OFF + t] = 0.0f;
    }
    if (t < 512) ws[BNSUM_OFF + t] = 0.0f;               /* 256 doubles */
}

/* ---------------- BatchNorm statistics (f64 accumulation) ---------------- */
__global__ __launch_bounds__(128) void k_bnstats(const float* __restrict__ x,
                                                 float* __restrict__ ws) {
    int f = threadIdx.x;                                 /* 128 features */
    float s = 0.f, s2 = 0.f;
    for (int row = blockIdx.x; row < N_NODES; row += gridDim.x) {
        float v = x[(size_t)row * F_IN + f];
        s += v; s2 += v * v;
    }
    double* bs = (double*)(ws + BNSUM_OFF);
    atomicAdd(&bs[f], (double)s);
    atomicAdd(&bs[128 + f], (double)s2);
}

__global__ __launch_bounds__(128) void k_bnfinalize(const float* __restrict__ gamma,
                                                    const float* __restrict__ beta,
                                                    float* __restrict__ ws) {
    int f = threadIdx.x;
    const double* bs = (const double*)(ws + BNSUM_OFF);
    double mean = bs[f] * (1.0 / N_NODES);
    double var  = bs[128 + f] * (1.0 / N_NODES) - mean * mean;
    float inv   = rsqrtf((float)var + BN_EPS);
    float sc    = gamma[f] * inv;
    ws[SCALE_OFF + f] = sc;
    ws[SHIFT_OFF + f] = beta[f] - (float)mean * sc;
}

/* -- transpose/concat weights, K-pair interleaved: wT2[(k/2)*512+n][k&1] --
 * This makes the (k, k+1) B-fragment pair for one output column 8 B
 * contiguous so the GEMM's B loads become single global_load_b64s.       */
__global__ __launch_bounds__(256) void k_prep(const float* __restrict__ w_l,
                                              const float* __restrict__ b_l,
                                              const float* __restrict__ w_r,
                                              const float* __restrict__ b_r,
                                              float* __restrict__ ws) {
    int t = blockIdx.x * 256 + threadIdx.x;              /* F_IN*OUTC = 65536 */
    if (t < F_IN * OUTC) {
        int k = t / OUTC, n = t % OUTC;
        float v = (n < HC) ? w_l[(size_t)n * F_IN + k]
                           : w_r[(size_t)(n - HC) * F_IN + k];
        ws[WT_OFF + (((size_t)(k >> 1) * OUTC + n) << 1) + (k & 1)] = v;
    }
    if (t < OUTC)
        ws[BIAS2_OFF + t] = (t < HC) ? b_l[t] : b_r[t - HC];
}

/* -------- fused BN-apply + dual GEMM via V_WMMA_F32_16X16X4_F32 ----------
 * Block: 16-row node tile, 8 waves x 4 column tiles cover N_out = 512.
 * A lane layout: M = lane%16, K = 2*(lane/16)+vgpr  (ISA 7.12.2, 32-bit A 16x4)
 * B lane layout: N = lane%16, K = 2*(lane/16)+vgpr  (mirror; lane = output col)
 * D lane layout: N = lane%16, M = r + 8*(lane/16), r = vgpr 0..7
 */
#define LDSTRIDE 130   /* pad 16x128 tile so column reads hit distinct banks */
__global__ __launch_bounds__(256) void k_gemm(const float* __restrict__ x,
                                              float* __restrict__ ws) {
    __shared__ float lds[16 * LDSTRIDE];
    const float* scale = ws + SCALE_OFF;
    const float* shift = ws + SHIFT_OFF;
    const float* wT2   = ws + WT_OFF;
    const float* bias2 = ws + BIAS2_OFF;
    float*       xsd   = ws + XSD_OFF;

    const int m0  = blockIdx.x * 16;
    const int tid = threadIdx.x;

    /* stage x tile with BN applied (coalesced: consecutive tid -> consecutive c) */
    #pragma unroll
    for (int i = 0; i < 8; ++i) {
        int id = i * 256 + tid;
        int r = id >> 7, c = id & 127;
        lds[r * LDSTRIDE + c] = x[(size_t)(m0 + r) * F_IN + c] * scale[c] + shift[c];
    }
    __syncthreads();

    const int lane  = tid & 31;
    const int wv    = tid >> 5;
    const int lrow  = lane & 15;          /* M for A, N for B/D */
    const int kgrp  = lane >> 4;          /* 0 or 1: which K pair of the 4   */
    const int khalf = kgrp << 1;          /* 0 or 2 */
    const int mbase = m0 + (kgrp << 3);

    for (int ct = wv * 4; ct < wv * 4 + 4; ++ct) {
        const int col0 = ct * 16;
        /* per-lane base into interleaved wT2 for this column */
        const v2f* bcol = (const v2f*)(wT2 + (((size_t)kgrp * OUTC + col0 + lrow) << 1));
        v8f acc = {};
        #pragma unroll
        for (int k0 = 0; k0 < F_IN; k0 += 4) {
            v2f a, b;
            a.x = lds[lrow * LDSTRIDE + k0 + khalf];
            a.y = lds[lrow * LDSTRIDE + k0 + khalf + 1];
            b   = bcol[(k0 >> 1) * OUTC];     /* one 8 B load: (k0+khalf, k0+khalf+1) */
            acc = __builtin_amdgcn_wmma_f32_16x16x4_f32(
                      false, a, false, b, (short)0, acc, false, false);
        }
        const float bb = bias2[col0 + lrow];
        #pragma unroll
        for (int r = 0; r < 8; ++r)
            xsd[(size_t)(mbase + r) * OUTC + col0 + lrow] = acc[r] + bb;
    }
}

/* -------- per-(edge,head) GATv2 logits + segment max (L2-resident gathers) */
__global__ __launch_bounds__(256) void k_logits(const int* __restrict__ ei,
                                                const float* __restrict__ att,
                                                float* __restrict__ ws) {
    int t = blockIdx.x * 256 + threadIdx.x;
    if (t >= E_TOT * HEADS) return;
    int e = t >> 2, h = t & 3;
    int src, dst;
    if (e < N_EDGES) { src = ei[e]; dst = ei[N_EDGES + e]; }
    else             { src = dst = e - N_EDGES; }

    const float* xsd = ws + XSD_OFF;
    const float4* xs = (const float4*)(xsd + (size_t)src * OUTC + h * HID);
    const float4* xd = (const float4*)(xsd + (size_t)dst * OUTC + HC + h * HID);
    const float4* a4 = (const float4*)(att + h * HID);

    float s = 0.f;
    #pragma unroll
    for (int i = 0; i < HID / 4; ++i) {
        float4 ms = xs[i], md = xd[i], av = a4[i];
        float v;
        v = ms.x + md.x; v = v > 0.f ? v : NEG_SLOPE * v; s += av.x * v;
        v = ms.y + md.y; v = v > 0.f ? v : NEG_SLOPE * v; s += av.y * v;
        v = ms.z + md.z; v = v > 0.f ? v : NEG_SLOPE * v; s += av.z * v;
        v = ms.w + md.w; v = v > 0.f ? v : NEG_SLOPE * v; s += av.w * v;
    }
    ws[P_OFF + (size_t)t] = s;
    atomicMax(&((unsigned*)(ws + SEGMAX_OFF))[(size_t)dst * HEADS + h], enc_f32(s));
}

/* -------- p = exp(logit - segmax), segment sum -------- */
__global__ __launch_bounds__(256) void k_softmax(const int* __restrict__ ei,
                                                 float* __restrict__ ws) {
    int t = blockIdx.x * 256 + threadIdx.x;
    if (t >= E_TOT * HEADS) return;
    int e = t >> 2, h = t & 3;
    int dst = (e < N_EDGES) ? ei[N_EDGES + e] : (e - N_EDGES);
    float m  = dec_f32(((const unsigned*)(ws + SEGMAX_OFF))[(size_t)dst * HEADS + h]);
    float pv = __expf(ws[P_OFF + (size_t)t] - m);
    ws[P_OFF + (size_t)t] = pv;
    atomicAdd(&ws[SEGSUM_OFF + (size_t)dst * HEADS + h], pv);
}

/* -------- scatter-aggregate alpha * xs[src] into acc[dst] -------- */
__global__ __launch_bounds__(256) void k_aggregate(const int* __restrict__ ei,
                                                   float* __restrict__ ws) {
    int t = blockIdx.x * 256 + threadIdx.x;
    if (t >= E_TOT * HEADS) return;
    int e = t >> 2, h = t & 3;
    int src, dst;
    if (e < N_EDGES) { src = ei[e]; dst = ei[N_EDGES + e]; }
    else             { src = dst = e - N_EDGES; }

    float alpha = ws[P_OFF + (size_t)t] /
                  ws[SEGSUM_OFF + (size_t)dst * HEADS + h];
    const float4* xs  = (const float4*)(ws + XSD_OFF + (size_t)src * OUTC + h * HID);
    float*        acc = ws + ACC_OFF + (size_t)dst * HC + h * HID;
    #pragma unroll
    for (int i = 0; i < HID / 4; ++i) {
        float4 v = xs[i];
        atomicAdd(&acc[4 * i + 0], alpha * v.x);
        atomicAdd(&acc[4 * i + 1], alpha * v.y);
        atomicAdd(&acc[4 * i + 2], alpha * v.z);
        atomicAdd(&acc[4 * i + 3], alpha * v.w);
    }
}

/* -------- head mean + bias + ELU -------- */
__global__ __launch_bounds__(256) void k_finalize(const float* __restrict__ ws,
                                                  const float* __restrict__ bias,
                                                  float* __restrict__ out) {
    int t = blockIdx.x * 256 + threadIdx.x;   /* N_NODES*HID */
    if (t >= N_NODES * HID) return;
    int n = t >> 6, c = t & 63;
    const float* a = ws + ACC_OFF + (size_t)n * HC;
    float v = (a[c] + a[64 + c] + a[128 + c] + a[192 + c]) * 0.25f + bias[c];
    out[t] = v > 0.f ? v : expm1f(v);
}

extern "C" void kernel_launch(void* const* d_in, const int* in_sizes, int n_in,
                              void* d_out, int out_size, void* d_ws, size_t ws_size,
                              hipStream_t stream) {
    const float* x     = (const float*)d_in[0];
    const int*   ei    = (const int*)  d_in[1];
    /* d_in[2] = batch, d_in[3] = num_graphs: shape-only, unused in math */
    const float* gamma = (const float*)d_in[4];
    const float* beta  = (const float*)d_in[5];
    const float* w_l   = (const float*)d_in[6];
    const float* b_l   = (const float*)d_in[7];
    const float* w_r   = (const float*)d_in[8];
    const float* b_r   = (const float*)d_in[9];
    const float* att   = (const float*)d_in[10];
    const float* bias  = (const float*)d_in[11];
    float* out = (float*)d_out;
    float* ws  = (float*)d_ws;   /* needs ~222 MB */
    (void)in_sizes; (void)n_in; (void)out_size; (void)ws_size;

    const int eh_blocks = (E_TOT * HEADS + 255) / 256;            /* 17408 */

    k_init      <<<(N_NODES * HC) / 256, 256, 0, stream>>>(ws);   /* 65536 blk */
    k_bnstats   <<<512, 128, 0, stream>>>(x, ws);
    k_bnfinalize<<<1, 128, 0, stream>>>(gamma, beta, ws);
    k_prep      <<<(F_IN * OUTC) / 256, 256, 0, stream>>>(w_l, b_l, w_r, b_r, ws);
    k_gemm      <<<N_NODES / 16, 256, 0, stream>>>(x, ws);        /* 4096 blk, WMMA */
    k_logits    <<<eh_blocks, 256, 0, stream>>>(ei, att, ws);
    k_softmax   <<<eh_blocks, 256, 0, stream>>>(ei, ws);
    k_aggregate <<<eh_blocks, 256, 0, stream>>>(ei, ws);
    k_finalize  <<<(N_NODES * HID) / 256, 256, 0, stream>>>(ws, bias, out);
}